// MyModel_49289044689230
// MI455X (gfx1250) — compile-verified
//
#include <hip/hip_runtime.h>

// GraphSAGE 2-layer forward for MI455X (gfx1250, wave32, WMMA + async-to-LDS).
//
//   deg  = segment_sum(1, dst);  inv_deg = 1/max(deg,1)            [atomics]
//   hA   = x @ W_t + b_t                                           [WMMA f32]
//   agg  = segment_sum(hA[src], dst)                               [f32 atomics, L2-resident]
//   hC   = relu(hA@W_self0 + b_self0 + inv_deg.*(agg@W_neigh0))    [WMMA f32, fused]
//   agg  = segment_sum(hC[src], dst)
//   hA   = relu(hC@W_self1 + b_self1 + inv_deg.*(agg@W_neigh1))
//   out  = hA @ W_fin                                              [wave dot]

#define HID 128
#define INF 256

typedef __attribute__((ext_vector_type(2))) float v2f;
typedef __attribute__((ext_vector_type(8))) float v8f;

// Async DMA: global -> LDS, 16 bytes, tracked by ASYNCcnt (no VGPR round-trip).
// VDST holds the LDS byte address (= low 32 bits of the flat shared pointer).
__device__ __forceinline__ void async_copy_b128(const float* gptr, float* lptr) {
  unsigned lds = (unsigned)(unsigned long long)lptr;
  unsigned long long ga = (unsigned long long)gptr;
  asm volatile("global_load_async_to_lds_b128 %0, %1, off"
               :
               : "v"(lds), "v"(ga)
               : "memory");
}
__device__ __forceinline__ void async_wait0() {
  asm volatile("s_wait_asynccnt 0x0" ::: "memory");
}

__global__ void k_zero(float* __restrict__ p, long n) {
  long i = (long)blockIdx.x * blockDim.x + threadIdx.x;
  if (i < n) p[i] = 0.0f;
}

__global__ void k_deg(const int* __restrict__ dst, float* __restrict__ deg, int nE) {
  int e = blockIdx.x * blockDim.x + threadIdx.x;
  if (e < nE) atomicAdd(&deg[dst[e]], 1.0f);
}

__global__ void k_invdeg(float* __restrict__ deg, int nN) {
  int i = blockIdx.x * blockDim.x + threadIdx.x;
  if (i < nN) deg[i] = 1.0f / fmaxf(deg[i], 1.0f);
}

// One wave (32 lanes) per edge: 128 floats = 4 per lane.
// Gather h[src] coalesced (b128), scatter-add into agg[dst] with native f32 atomics.
__global__ __launch_bounds__(256) void k_scatter(const float* __restrict__ h,
                                                 const int* __restrict__ src,
                                                 const int* __restrict__ dst,
                                                 float* __restrict__ agg, int nE) {
  int wid  = (int)((blockIdx.x * blockDim.x + threadIdx.x) >> 5);
  int lane = threadIdx.x & 31;
  if (wid >= nE) return;
  long s = (long)src[wid] * HID;
  long d = (long)dst[wid] * HID;
  float4 v = *(const float4*)&h[s + lane * 4];
  atomicAdd(&agg[d + lane * 4 + 0], v.x);
  atomicAdd(&agg[d + lane * 4 + 1], v.y);
  atomicAdd(&agg[d + lane * 4 + 2], v.z);
  atomicAdd(&agg[d + lane * 4 + 3], v.w);
}

// C[N x 128] = act( A[N x K] @ W[K x 128] + bias  (+ scale[row] * (A2 @ W2)[row,:]) )
//
// Block: 256 threads = 8 waves; block owns a 32-row strip, wave w owns cols [16w,16w+16)
// with two 16x16 accumulators per operand pair (rows 0-15 / 16-31), so each B fragment
// feeds 2 WMMAs. A strips go global->LDS via GLOBAL_LOAD_ASYNC_TO_LDS_B128 (ASYNCcnt),
// row stride K+4 floats (conflict-free ds_load_b64 fragments, 16B-aligned rows).
// Mean-normalization is applied per-row at the epilogue: each C accumulator element
// belongs to exactly one row, so  relu(self + inv_deg[r]*neigh + b)  is exact.
//
// V_WMMA_F32_16X16X4_F32 fragment mapping (wave32):
//   A (16x4):  {half=lane>>4, m=lane&15}: a.x=A[m][k+2*half], a.y=A[m][k+2*half+1]
//   B (4x16):  b.x=W[k+2*half][col0+m],  b.y=W[k+1+2*half][col0+m]
//   C (16x16): acc[v] -> row (v + 8*half), col (col0 + m)
template <int K, bool SECOND, bool RELU>
__global__ __launch_bounds__(256) void k_gemm(const float* __restrict__ A,
                                              const float* __restrict__ W,
                                              const float* __restrict__ bias,
                                              const float* __restrict__ A2,
                                              const float* __restrict__ W2,
                                              const float* __restrict__ scale,
                                              float* __restrict__ out, int N) {
  constexpr int KP = K + 4;  // pad to dodge 16-way LDS bank conflicts
  __shared__ float sA[32 * KP];
  __shared__ float sB[SECOND ? 32 * KP : 1];
  __shared__ float sSc[SECOND ? 32 : 1];

  const int tid  = threadIdx.x;
  const int row0 = blockIdx.x * 32;
  const bool full = (row0 + 32 <= N);  // uniform per block

  if (full) {
    // Async DMA staging: L2 -> LDS, 16B granules, no VGPR round-trip.
    constexpr int NV4 = 32 * K / 4;
    for (int idx = tid; idx < NV4; idx += 256) {
      int r  = idx / (K / 4);
      int c4 = (idx - r * (K / 4)) * 4;
      async_copy_b128(&A[(long)(row0 + r) * K + c4], &sA[r * KP + c4]);
      if constexpr (SECOND)
        async_copy_b128(&A2[(long)(row0 + r) * K + c4], &sB[r * KP + c4]);
    }
    if constexpr (SECOND)
      if (tid < 32) sSc[tid] = scale[row0 + tid];
    async_wait0();  // each wave drains its own ASYNCcnt; barrier covers the rest
  } else {
    for (int idx = tid; idx < 32 * K; idx += 256) {
      int r = idx / K, c = idx - r * K;
      int gr = row0 + r;
      sA[r * KP + c] = (gr < N) ? A[(long)gr * K + c] : 0.0f;
      if constexpr (SECOND)
        sB[r * KP + c] = (gr < N) ? A2[(long)gr * K + c] : 0.0f;
    }
    if constexpr (SECOND)
      if (tid < 32) sSc[tid] = (row0 + tid < N) ? scale[row0 + tid] : 0.0f;
  }
  __syncthreads();

  const int wave    = tid >> 5;
  const int lane    = tid & 31;
  const int half    = lane >> 4;
  const int m       = lane & 15;
  const int colbase = wave * 16;
  const int aoff    = m * KP + 2 * half;         // fragment base, rows 0-15
  const int aoff2   = (m + 16) * KP + 2 * half;  // fragment base, rows 16-31

  v8f acc0 = {}, acc1 = {};
#pragma unroll 4
  for (int k = 0; k < K; k += 4) {
    v2f b;
    b.x = W[(long)(k + 2 * half) * HID + colbase + m];
    b.y = W[(long)(k + 1 + 2 * half) * HID + colbase + m];
    v2f a0 = *(const v2f*)&sA[aoff + k];
    v2f a1 = *(const v2f*)&sA[aoff2 + k];
    acc0 = __builtin_amdgcn_wmma_f32_16x16x4_f32(false, a0, false, b, (short)0, acc0,
                                                 false, false);
    acc1 = __builtin_amdgcn_wmma_f32_16x16x4_f32(false, a1, false, b, (short)0, acc1,
                                                 false, false);
  }

  v8f accN0 = {}, accN1 = {};
  if constexpr (SECOND) {
#pragma unroll 4
    for (int k = 0; k < K; k += 4) {
      v2f b;
      b.x = W2[(long)(k + 2 * half) * HID + colbase + m];
      b.y = W2[(long)(k + 1 + 2 * half) * HID + colbase + m];
      v2f a0 = *(const v2f*)&sB[aoff + k];
      v2f a1 = *(const v2f*)&sB[aoff2 + k];
      accN0 = __builtin_amdgcn_wmma_f32_16x16x4_f32(false, a0, false, b, (short)0,
                                                    accN0, false, false);
      accN1 = __builtin_amdgcn_wmma_f32_16x16x4_f32(false, a1, false, b, (short)0,
                                                    accN1, false, false);
    }
  }

  const int nc    = colbase + m;
  const float bv  = bias[nc];
  const int rloc  = 8 * half;        // local row base within the 32-row strip
  const int rbase = row0 + rloc;

  if (full) {
#pragma unroll
    for (int v = 0; v < 8; ++v) {
      float val = acc0[v] + bv;
      if constexpr (SECOND) val += sSc[rloc + v] * accN0[v];
      if constexpr (RELU) val = fmaxf(val, 0.0f);
      out[(long)(rbase + v) * HID + nc] = val;

      float val1 = acc1[v] + bv;
      if constexpr (SECOND) val1 += sSc[16 + rloc + v] * accN1[v];
      if constexpr (RELU) val1 = fmaxf(val1, 0.0f);
      out[(long)(rbase + 16 + v) * HID + nc] = val1;
    }
  } else {
#pragma unroll
    for (int v = 0; v < 8; ++v) {
      int r = rbase + v;
      if (r < N) {
        float val = acc0[v] + bv;
        if constexpr (SECOND) val += sSc[rloc + v] * accN0[v];
        if constexpr (RELU) val = fmaxf(val, 0.0f);
        out[(long)r * HID + nc] = val;
      }
      int r2 = rbase + 16 + v;
      if (r2 < N) {
        float val1 = acc1[v] + bv;
        if constexpr (SECOND) val1 += sSc[16 + rloc + v] * accN1[v];
        if constexpr (RELU) val1 = fmaxf(val1, 0.0f);
        out[(long)r2 * HID + nc] = val1;
      }
    }
  }
}

// out[n] = dot(h[n, 0:128], W_fin[0:128]) — one wave per node, shfl reduction.
__global__ __launch_bounds__(256) void k_fin(const float* __restrict__ h,
                                             const float* __restrict__ wfin,
                                             float* __restrict__ out, int N) {
  int wid  = (int)((blockIdx.x * blockDim.x + threadIdx.x) >> 5);
  int lane = threadIdx.x & 31;
  if (wid >= N) return;
  float4 v = *(const float4*)&h[(long)wid * HID + lane * 4];
  float4 w = *(const float4*)&wfin[lane * 4];
  float s = v.x * w.x + v.y * w.y + v.z * w.z + v.w * w.w;
#pragma unroll
  for (int off = 16; off > 0; off >>= 1) s += __shfl_down(s, off, 32);
  if (lane == 0) out[wid] = s;
}

extern "C" void kernel_launch(void* const* d_in, const int* in_sizes, int n_in,
                              void* d_out, int out_size, void* d_ws, size_t ws_size,
                              hipStream_t stream) {
  const float* x     = (const float*)d_in[0];
  const float* W_t   = (const float*)d_in[1];
  const float* b_t   = (const float*)d_in[2];
  const float* W_s0  = (const float*)d_in[3];
  const float* b_s0  = (const float*)d_in[4];
  const float* W_n0  = (const float*)d_in[5];
  const float* W_s1  = (const float*)d_in[6];
  const float* b_s1  = (const float*)d_in[7];
  const float* W_n1  = (const float*)d_in[8];
  const float* W_fin = (const float*)d_in[9];
  const int*   src   = (const int*)d_in[10];
  const int*   dst   = (const int*)d_in[11];
  float* out = (float*)d_out;

  const int N = in_sizes[0] / INF;  // 100000
  const int E = in_sizes[10];       // 1600000

  // Workspace layout (floats): hA | agg | hC | deg   (~154 MB)
  float* hA  = (float*)d_ws;
  float* agg = hA + (size_t)N * HID;
  float* hC  = agg + (size_t)N * HID;
  float* deg = hC + (size_t)N * HID;

  const long zn = (long)N * HID;
  const int  gemmBlocks = (N + 31) / 32;
  const int  edgeWaveBlocks = (E * 32 + 255) / 256;

  // Degrees (shared by both layers) and first agg buffer.
  k_zero<<<(unsigned)((zn + 255) / 256), 256, 0, stream>>>(agg, zn);
  k_zero<<<(N + 255) / 256, 256, 0, stream>>>(deg, N);
  k_deg<<<(E + 255) / 256, 256, 0, stream>>>(dst, deg, E);
  k_invdeg<<<(N + 255) / 256, 256, 0, stream>>>(deg, N);  // deg now holds 1/max(deg,1)

  // hA = x @ W_t + b_t
  k_gemm<INF, false, false><<<gemmBlocks, 256, 0, stream>>>(
      x, W_t, b_t, nullptr, nullptr, nullptr, hA, N);

  // Layer 0
  k_scatter<<<edgeWaveBlocks, 256, 0, stream>>>(hA, src, dst, agg, E);
  k_gemm<HID, true, true><<<gemmBlocks, 256, 0, stream>>>(
      hA, W_s0, b_s0, agg, W_n0, deg, hC, N);

  // Layer 1
  k_zero<<<(unsigned)((zn + 255) / 256), 256, 0, stream>>>(agg, zn);
  k_scatter<<<edgeWaveBlocks, 256, 0, stream>>>(hC, src, dst, agg, E);
  k_gemm<HID, true, true><<<gemmBlocks, 256, 0, stream>>>(
      hC, W_s1, b_s1, agg, W_n1, deg, hA, N);

  // out = hA @ W_fin
  k_fin<<<(N * 32 + 255) / 256, 256, 0, stream>>>(hA, W_fin, out, N);
}